// GumbelVectorQuantizer_8521215115482
// MI455X (gfx1250) — compile-verified
//
#include <hip/hip_runtime.h>
#include <stdint.h>

typedef __attribute__((ext_vector_type(2))) float v2f;
typedef __attribute__((ext_vector_type(4))) float v4f;
typedef __attribute__((ext_vector_type(8))) float v8f;
typedef __attribute__((ext_vector_type(4))) unsigned int v4u;
typedef __attribute__((ext_vector_type(4))) int v4i;
typedef __attribute__((ext_vector_type(8))) int v8i;

#define BT_TOT 64000
#define FIN    1024
#define GV     640   // GROUPS*NUM_VARS
#define NV     320
#define VD     128
#define KC     256   // K chunk staged in LDS by the TDM
#define MT     16    // rows per workgroup
#define EPS_F  1e-7f

// ---------------------------------------------------------------------------
// Tensor Data Mover: load a [MT rows x KC cols] fp32 tile (row stride FIN)
// from global into LDS at byte offset lds_off.  Descriptor per CDNA5 ISA §8.
// This toolchain exposes the 6-arg builtin (g0, g1, g2, g3, extra, cpol).
// ---------------------------------------------------------------------------
__device__ __forceinline__ void tdm_load_tile(const float* gptr, unsigned lds_off) {
    unsigned long long ga = (unsigned long long)(uintptr_t)gptr;
    v4u g0;
    g0.x = 1u;                                           // count=1, user desc
    g0.y = lds_off;                                      // lds_addr (bytes)
    g0.z = (unsigned)(ga & 0xFFFFFFFFu);                 // global_addr[31:0]
    g0.w = (unsigned)((ga >> 32) & 0x1FFFFFFu)           // global_addr[56:32]
         | (2u << 30);                                   // type=2 ("image")
    v8i g1;
    g1[0] = (int)(2u << 16);                             // data_size=2 (4 B), wg_mask=0
    g1[1] = (int)((unsigned)KC << 16);                   // tensor_dim0[15:0] = KC
    g1[2] = (int)((unsigned)MT << 16);                   // dim0 hi=0 ; tensor_dim1[15:0]=MT
    g1[3] = (int)((unsigned)KC << 16);                   // dim1 hi=0 ; tile_dim0=KC
    g1[4] = (int)MT;                                     // tile_dim1=MT ; tile_dim2=0
    g1[5] = (int)FIN;                                    // tensor_dim0_stride lo32 = FIN
    g1[6] = 0;                                           // stride hi / dim1_stride lo
    g1[7] = 0;
    v4i z4 = (v4i){0, 0, 0, 0};
    v8i z8 = (v8i){0, 0, 0, 0, 0, 0, 0, 0};
    __builtin_amdgcn_tensor_load_to_lds(g0, g1, z4, z4, z8, 0);
}

// ---------------------------------------------------------------------------
// Fused kernel: GEMM (f32 WMMA) + softmax/argmax stats + codebook gather.
// Grid: BT_TOT/MT blocks of 256 threads (8 wave32).
// Dynamic LDS layout (floats):
//   xs     [2*MT*KC]   double-buffered x tile (TDM destination)
//   lgs    [MT*GV]     logits for the 16 rows
//   hp     [GV]        per-WG hard-count partial
//   ap     [GV]        per-WG softmax-prob partial
// ---------------------------------------------------------------------------
__global__ void __launch_bounds__(256)
gvq_main(const float* __restrict__ x, const float* __restrict__ w,
         const float* __restrict__ bias, const float* __restrict__ cb,
         const float* __restrict__ gum, float* __restrict__ out,
         float* __restrict__ ws_hard, float* __restrict__ ws_avg)
{
    extern __shared__ float smem[];
    float* xs  = smem;                       // 2*MT*KC = 8192
    float* lgs = smem + 2 * MT * KC;         // MT*GV   = 10240
    float* hp  = lgs + MT * GV;              // GV
    float* ap  = hp + GV;                    // GV

    const int tid  = threadIdx.x;
    const int lane = tid & 31;
    const int wave = tid >> 5;
    const int row0 = blockIdx.x * MT;

    for (int i = tid; i < GV; i += 256) { hp[i] = 0.0f; ap[i] = 0.0f; }

    const unsigned lds_base = (unsigned)(uintptr_t)xs;
    const bool leader = (__builtin_amdgcn_readfirstlane(wave) == 0);

    if (leader) {
        tdm_load_tile(x + (size_t)row0 * FIN, lds_base);
        __builtin_amdgcn_s_wait_tensorcnt(0);
    }
    __syncthreads();

    // ---- GEMM: each wave computes 5 N-tiles of 16 columns, M=16, K=1024 ----
    v8f acc[5];
#pragma unroll
    for (int t = 0; t < 5; ++t) acc[t] = (v8f){0, 0, 0, 0, 0, 0, 0, 0};

    const int mrow  = lane & 15;        // M / N index within tile
    const int khalf = (lane >> 4) << 1; // K sub-offset {0,2} per half-wave

    const float* wbase[5];
#pragma unroll
    for (int t = 0; t < 5; ++t) {
        int n = (wave * 5 + t) * 16 + mrow;
        wbase[t] = w + (size_t)n * FIN + khalf;   // B fragment: W[n][k..]
    }

    const int NCHUNK = FIN / KC;
    for (int c = 0; c < NCHUNK; ++c) {
        if (leader && (c + 1) < NCHUNK)
            tdm_load_tile(x + (size_t)row0 * FIN + (size_t)(c + 1) * KC,
                          lds_base + (unsigned)(((c + 1) & 1) * MT * KC * 4));

        const float* xa = xs + (c & 1) * MT * KC + mrow * KC + khalf;
        const int kbase = c * KC;
        for (int kk = 0; kk < KC; kk += 4) {
            v2f a = *(const v2f*)(xa + kk);             // ds_load_b64
#pragma unroll
            for (int t = 0; t < 5; ++t) {
                v2f b = *(const v2f*)(wbase[t] + kbase + kk);  // global_load_b64 (L2-hot)
                acc[t] = __builtin_amdgcn_wmma_f32_16x16x4_f32(
                    false, a, false, b, (short)0, acc[t], false, false);
            }
        }
        if (leader && (c + 1) < NCHUNK) __builtin_amdgcn_s_wait_tensorcnt(0);
        __syncthreads();
    }

    // ---- spill logits (+bias) to LDS per C-layout: M = j + 8*(lane>>4), N = lane&15
    const int mh = (lane >> 4) * 8;
#pragma unroll
    for (int t = 0; t < 5; ++t) {
        int nn = (wave * 5 + t) * 16 + mrow;
        float bv = bias[nn];
#pragma unroll
        for (int j = 0; j < 8; ++j)
            lgs[(mh + j) * GV + nn] = acc[t][j] + bv;
    }
    __syncthreads();

    // ---- phase 2: per (row, group) softmax / argmax / gumbel-argmax ----
    for (int rr = 0; rr < 2; ++rr) {
        const int r = wave * 2 + rr;
        const size_t grow = (size_t)(row0 + r) * GV;
        for (int g = 0; g < 2; ++g) {
            const float* lr = lgs + r * GV + g * NV;
            float vals[10];
            float m = -3.4e38f; int am = 0;
#pragma unroll
            for (int i = 0; i < 10; ++i) {
                int idx = lane + 32 * i;
                float v = lr[idx];
                vals[i] = v;
                if (v > m) { m = v; am = idx; }
            }
            const float* gp = gum + grow + g * NV;
            float gm = -3.4e38f; int gam = 0;
#pragma unroll
            for (int i = 0; i < 10; ++i) {
                int idx = lane + 32 * i;
                float v = vals[i] + gp[idx];
                if (v > gm) { gm = v; gam = idx; }
            }
            // wave32 butterfly argmax reductions (first-index tie-break)
#pragma unroll
            for (int off = 16; off >= 1; off >>= 1) {
                float om = __shfl_xor(m, off, 32);
                int   oa = __shfl_xor(am, off, 32);
                if (om > m || (om == m && oa < am)) { m = om; am = oa; }
                float og = __shfl_xor(gm, off, 32);
                int   ob = __shfl_xor(gam, off, 32);
                if (og > gm || (og == gm && ob < gam)) { gm = og; gam = ob; }
            }
            // softmax of raw logits -> avg_probs partial
            float s = 0.0f;
#pragma unroll
            for (int i = 0; i < 10; ++i) { vals[i] = __expf(vals[i] - m); s += vals[i]; }
#pragma unroll
            for (int off = 16; off >= 1; off >>= 1) s += __shfl_xor(s, off, 32);
            float inv = 1.0f / s;
#pragma unroll
            for (int i = 0; i < 10; ++i)
                atomicAdd(&ap[g * NV + lane + 32 * i], vals[i] * inv);
            if (lane == 0) atomicAdd(&hp[g * NV + am], 1.0f);

            // q = codebook[g*NV + argmax(logits+gumbel)]  (straight-through == hard)
            const float* crow = cb + (size_t)(g * NV + gam) * VD;
            float* orow = out + (size_t)(row0 + r) * 256 + g * VD;
            *(v4f*)(orow + lane * 4) = *(const v4f*)(crow + lane * 4);
        }
    }
    __syncthreads();

    for (int i = tid; i < GV; i += 256) {
        atomicAdd(&ws_hard[i], hp[i]);
        atomicAdd(&ws_avg[i], ap[i]);
    }
}

// ---------------------------------------------------------------------------
__global__ void gvq_zero(float* ws, int n) {
    for (int i = threadIdx.x; i < n; i += 256) ws[i] = 0.0f;
}

__global__ void __launch_bounds__(GV)
gvq_final(const float* __restrict__ ws_hard, const float* __restrict__ ws_avg,
          float* __restrict__ out)
{
    __shared__ float sh[GV];
    __shared__ float sa[GV];
    const int i = threadIdx.x;
    const float scl = 1.0f / (float)BT_TOT;
    float h = ws_hard[i] * scl;
    float a = ws_avg[i] * scl;
    sh[i] = h * __logf(h + EPS_F);
    sa[i] = a * __logf(a + EPS_F);
    __syncthreads();
    if (i == 0) {
        float h0 = 0, h1 = 0, a0 = 0, a1 = 0;
        for (int v = 0; v < NV; ++v) {
            h0 += sh[v]; h1 += sh[NV + v];
            a0 += sa[v]; a1 += sa[NV + v];
        }
        out[16384000] = __expf(-h0) + __expf(-h1);  // code_perplexity
        out[16384001] = __expf(-a0) + __expf(-a1);  // prob_perplexity
    }
}

// ---------------------------------------------------------------------------
extern "C" void kernel_launch(void* const* d_in, const int* in_sizes, int n_in,
                              void* d_out, int out_size, void* d_ws, size_t ws_size,
                              hipStream_t stream) {
    const float* x   = (const float*)d_in[0];
    const float* w   = (const float*)d_in[1];
    const float* b   = (const float*)d_in[2];
    const float* cb  = (const float*)d_in[3];
    const float* gum = (const float*)d_in[4];
    float* out = (float*)d_out;
    float* ws  = (float*)d_ws;

    gvq_zero<<<1, 256, 0, stream>>>(ws, 2 * GV);

    const size_t shmem = (size_t)(2 * MT * KC + MT * GV + 2 * GV) * sizeof(float);
    gvq_main<<<BT_TOT / MT, 256, shmem, stream>>>(x, w, b, cb, gum, out, ws, ws + GV);

    gvq_final<<<1, GV, 0, stream>>>(ws, ws + GV, out);
}